// MultiGAT_local_53807350284443
// MI455X (gfx1250) — compile-verified
//
#include <hip/hip_runtime.h>
#include <math.h>

typedef float v2f __attribute__((ext_vector_type(2)));
typedef float v8f __attribute__((ext_vector_type(8)));

#define FIN 128
#define HEADS 2

// ---------------------------------------------------------------------------
// FP32 WMMA GEMM: Y[nrows,NCOLS] = X[nrows,128] @ W[128,NCOLS]
// One wave -> one 16x16 output tile via V_WMMA_F32_16X16X4_F32, K = 0..124.
// NCOLS is compile-time so all W/Y addressing folds into immediate offsets.
// A layout (ISA 7.12.2, 32-bit A 16x4): lanes 0-15 = M, VGPR0/1 = K0/K1;
// lanes 16-31 = K2/K3  -> contiguous float2 load per lane.
// ---------------------------------------------------------------------------
template <int NCOLS>
__global__ __launch_bounds__(32 * (NCOLS / 16)) void gat_gemm_wmma(
        const float* __restrict__ X,
        const float* __restrict__ W,
        float* __restrict__ Y,
        int nrows) {
    const int lane    = threadIdx.x & 31;
    const int wave    = threadIdx.x >> 5;
    const int rowBase = blockIdx.x * 16;
    const int colBase = wave * 16;
    const int m       = lane & 15;
    const int hi      = lane >> 4;          // 0 -> K pair {0,1}, 1 -> {2,3}

    int arow = rowBase + m;
    if (arow >= nrows) arow = nrows - 1;    // clamp (tail safety; stores guarded)
    const int col = colBase + m;

    // fixed per-lane bases; loop offsets are compile-time immediates
    const float* Xp = X + (size_t)arow * FIN + hi * 2;   // A: {k, k+1} contiguous
    const float* Wp = W + (size_t)hi * 2 * NCOLS + col;  // B: stride NCOLS rows

    __builtin_prefetch(Wp, 0, 1);                        // global_prefetch_b8

    v8f acc = {};
    #pragma unroll
    for (int k0 = 0; k0 < FIN; k0 += 4) {
        v2f a = *(const v2f*)(Xp + k0);
        v2f b;
        b.x = Wp[(size_t)k0 * NCOLS];
        b.y = Wp[(size_t)(k0 + 1) * NCOLS];
        acc = __builtin_amdgcn_wmma_f32_16x16x4_f32(
                  false, a, false, b, (short)0, acc, false, false);
    }

    float* Yp = Y + (size_t)(rowBase + hi * 8) * NCOLS + col;
    #pragma unroll
    for (int i = 0; i < 8; ++i) {
        if (rowBase + hi * 8 + i < nrows) Yp[(size_t)i * NCOLS] = acc[i];
    }
}

// ---------------------------------------------------------------------------
// Per-node attention coefficients + init of segment max/sum buffers.
// idx = n*2 + h
// ---------------------------------------------------------------------------
__global__ void gat_node_prep(const float* __restrict__ hlin,
                              const float* __restrict__ a_src,
                              const float* __restrict__ a_dst,
                              float* __restrict__ asb, float* __restrict__ adb,
                              float* __restrict__ mbuf, float* __restrict__ sbuf,
                              int nodes, int C) {
    int idx = blockIdx.x * blockDim.x + threadIdx.x;
    if (idx >= nodes * HEADS) return;
    int n = idx >> 1, h = idx & 1;
    const float* hp  = hlin  + (size_t)n * HEADS * C + (size_t)h * C;
    const float* asp = a_src + (size_t)h * C;
    const float* adp = a_dst + (size_t)h * C;
    float s0 = 0.f, s1 = 0.f;
    for (int c = 0; c < C; ++c) { float v = hp[c]; s0 += v * asp[c]; s1 += v * adp[c]; }
    asb[idx]  = s0;
    adb[idx]  = s1;
    mbuf[idx] = -__builtin_inff();
    sbuf[idx] = 0.f;
}

__device__ __forceinline__ void atomicMaxF(float* addr, float val) {
    // sign-split trick == GLOBAL_ATOMIC_MAX_NUM_F32 semantics (init = -inf)
    if (val >= 0.f) atomicMax((int*)addr, __float_as_int(val));
    else            atomicMin((unsigned int*)addr, __float_as_uint(val));
}

__device__ __forceinline__ void edge_ends(const int* __restrict__ ei, int e, int E,
                                          int& src, int& dst) {
    if (e < E) { src = ei[e]; dst = ei[E + e]; }
    else       { src = e - E; dst = e - E; }        // self loops appended
}

// Pass 1: e = leaky_relu(as[src]+ad[dst]); store; segment-max into m[dst]
__global__ void gat_edge_logits(const int* __restrict__ ei,
                                const float* __restrict__ asb,
                                const float* __restrict__ adb,
                                float* __restrict__ ebuf, float* __restrict__ mbuf,
                                int E, int nodes) {
    int e = blockIdx.x * blockDim.x + threadIdx.x;
    if (e >= E + nodes) return;
    int src, dst; edge_ends(ei, e, E, src, dst);
    #pragma unroll
    for (int h = 0; h < HEADS; ++h) {
        float v = asb[src * HEADS + h] + adb[dst * HEADS + h];
        v = (v > 0.f) ? v : 0.2f * v;               // NEG_SLOPE
        ebuf[(size_t)e * HEADS + h] = v;
        atomicMaxF(&mbuf[dst * HEADS + h], v);
    }
}

// Pass 2: ex = exp(e - m[dst]); store; segment-sum into s[dst]
__global__ void gat_edge_exp(const int* __restrict__ ei,
                             float* __restrict__ ebuf,
                             const float* __restrict__ mbuf,
                             float* __restrict__ sbuf,
                             int E, int nodes) {
    int e = blockIdx.x * blockDim.x + threadIdx.x;
    if (e >= E + nodes) return;
    int src, dst; edge_ends(ei, e, E, src, dst);
    (void)src;
    #pragma unroll
    for (int h = 0; h < HEADS; ++h) {
        float ex = expf(ebuf[(size_t)e * HEADS + h] - mbuf[dst * HEADS + h]);
        ebuf[(size_t)e * HEADS + h] = ex;
        atomicAdd(&sbuf[dst * HEADS + h], ex);
    }
}

// Pass 3: agg[dst, j] += h[src, j] * ex/(s[dst]+eps); thread per (edge, channel)
__global__ void gat_edge_agg(const int* __restrict__ ei,
                             const float* __restrict__ ebuf,
                             const float* __restrict__ sbuf,
                             const float* __restrict__ hlin,
                             float* __restrict__ agg,
                             int E, int nodes, int log2HC, int log2C) {
    long long idx = (long long)blockIdx.x * blockDim.x + threadIdx.x;
    long long tot = (long long)(E + nodes) << log2HC;
    if (idx >= tot) return;
    int e = (int)(idx >> log2HC);
    int j = (int)idx & ((1 << log2HC) - 1);
    int h = j >> log2C;
    int src, dst; edge_ends(ei, e, E, src, dst);
    float alpha = ebuf[(size_t)e * HEADS + h] / (sbuf[dst * HEADS + h] + 1e-16f);
    atomicAdd(&agg[((size_t)dst << log2HC) + j],
              hlin[((size_t)src << log2HC) + j] * alpha);
}

// bias + BatchNorm(eval) + ReLU, in place
__global__ void gat_bias_bn_relu(float* __restrict__ x,
                                 const float* __restrict__ b,
                                 const float* __restrict__ g,
                                 const float* __restrict__ be,
                                 const float* __restrict__ mu,
                                 const float* __restrict__ var,
                                 long long total, int F) {
    long long i = (long long)blockIdx.x * blockDim.x + threadIdx.x;
    if (i >= total) return;
    int j = (int)(i % F);
    float v = x[i] + b[j];
    v = (v - mu[j]) * rsqrtf(var[j] + 1e-5f) * g[j] + be[j];
    x[i] = fmaxf(v, 0.f);
}

// Layer-2 epilogue: head-mean + bias + BN, scatter into per-graph pools
__global__ void gat_finalize_pool(const float* __restrict__ agg2,
                                  const float* __restrict__ b2,
                                  const float* __restrict__ g,
                                  const float* __restrict__ be,
                                  const float* __restrict__ mu,
                                  const float* __restrict__ var,
                                  const int* __restrict__ batch,
                                  float* __restrict__ psum, float* __restrict__ pcnt,
                                  int nodes) {
    long long idx = (long long)blockIdx.x * blockDim.x + threadIdx.x;
    if (idx >= (long long)nodes * 32) return;
    int n = (int)(idx >> 5), c = (int)idx & 31;
    float v = 0.5f * (agg2[(size_t)n * 64 + c] + agg2[(size_t)n * 64 + 32 + c]) + b2[c];
    v = (v - mu[c]) * rsqrtf(var[c] + 1e-5f) * g[c] + be[c];
    int grp = batch[n];
    atomicAdd(&psum[(size_t)grp * 32 + c], v);
    if (c == 0) atomicAdd(&pcnt[grp], 1.f);
}

// One wave32 per graph: 32 lanes == 32 classes; mean pool + log_softmax
__global__ void gat_pool_logsoftmax(const float* __restrict__ psum,
                                    const float* __restrict__ pcnt,
                                    float* __restrict__ out, int G) {
    int grp = blockIdx.x;
    int c   = threadIdx.x;                 // 0..31
    float v = psum[(size_t)grp * 32 + c] / fmaxf(pcnt[grp], 1.f);
    float mx = v;
    #pragma unroll
    for (int off = 16; off > 0; off >>= 1) mx = fmaxf(mx, __shfl_xor(mx, off, 32));
    float ex = expf(v - mx);
    float sm = ex;
    #pragma unroll
    for (int off = 16; off > 0; off >>= 1) sm += __shfl_xor(sm, off, 32);
    out[(size_t)grp * 32 + c] = v - mx - logf(sm);
}

__global__ void gat_zero(float* __restrict__ p, long long n) {
    long long i = (long long)blockIdx.x * blockDim.x + threadIdx.x;
    if (i < n) p[i] = 0.f;
}

// ---------------------------------------------------------------------------
extern "C" void kernel_launch(void* const* d_in, const int* in_sizes, int n_in,
                              void* d_out, int out_size, void* d_ws, size_t ws_size,
                              hipStream_t stream) {
    const float* x     = (const float*)d_in[0];
    const int*   ei    = (const int*)  d_in[1];
    const int*   batch = (const int*)  d_in[2];
    const float* W1    = (const float*)d_in[3];
    const float* as1   = (const float*)d_in[4];
    const float* ad1   = (const float*)d_in[5];
    const float* b1    = (const float*)d_in[6];
    const float* g1    = (const float*)d_in[7];
    const float* be1   = (const float*)d_in[8];
    const float* mu1   = (const float*)d_in[9];
    const float* var1  = (const float*)d_in[10];
    const float* W2    = (const float*)d_in[11];
    const float* as2   = (const float*)d_in[12];
    const float* ad2   = (const float*)d_in[13];
    const float* b2    = (const float*)d_in[14];
    const float* g2    = (const float*)d_in[15];
    const float* be2   = (const float*)d_in[16];
    const float* mu2   = (const float*)d_in[17];
    const float* var2  = (const float*)d_in[18];

    const int N    = in_sizes[0] / FIN;       // 50000
    const int E    = in_sizes[1] / 2;         // 800000
    const int Etot = E + N;                   // self loops appended
    const int G    = out_size / 32;           // 512

    // workspace carve-up (floats)
    float* ws = (float*)d_ws;
    size_t off = 0;
    float* hlin1 = ws + off; off += (size_t)N * 128;
    float* agg1  = ws + off; off += (size_t)N * 128;
    float* hlin2 = ws + off; off += (size_t)N * 64;
    float* agg2  = ws + off; off += (size_t)N * 64;
    float* ebuf  = ws + off; off += (size_t)Etot * 2;
    float* mbuf  = ws + off; off += (size_t)N * 2;
    float* sbuf  = ws + off; off += (size_t)N * 2;
    float* asb   = ws + off; off += (size_t)N * 2;
    float* adb   = ws + off; off += (size_t)N * 2;
    float* psum  = ws + off; off += (size_t)G * 32;
    float* pcnt  = ws + off; off += (size_t)G;

    const int TB = 256;
    auto blocks = [](long long n, int tb) { return (unsigned)((n + tb - 1) / tb); };
    const int mStripes = (N + 15) / 16;
    const unsigned edgeBlk = blocks(Etot, TB);

    // ---------------- layer 1 (FIN=128 -> H*CH=128, concat) ----------------
    gat_zero<<<blocks((long long)N * 128, TB), TB, 0, stream>>>(agg1, (long long)N * 128);
    gat_gemm_wmma<128><<<mStripes, 32 * (128 / 16), 0, stream>>>(x, W1, hlin1, N);
    gat_node_prep<<<blocks((long long)N * 2, TB), TB, 0, stream>>>(
        hlin1, as1, ad1, asb, adb, mbuf, sbuf, N, 64);
    gat_edge_logits<<<edgeBlk, TB, 0, stream>>>(ei, asb, adb, ebuf, mbuf, E, N);
    gat_edge_exp<<<edgeBlk, TB, 0, stream>>>(ei, ebuf, mbuf, sbuf, E, N);
    gat_edge_agg<<<blocks((long long)Etot * 128, TB), TB, 0, stream>>>(
        ei, ebuf, sbuf, hlin1, agg1, E, N, 7 /*log2(128)*/, 6 /*log2(64)*/);
    gat_bias_bn_relu<<<blocks((long long)N * 128, TB), TB, 0, stream>>>(
        agg1, b1, g1, be1, mu1, var1, (long long)N * 128, 128);

    // ---------------- layer 2 (128 -> COUT=32, head-mean) ------------------
    gat_zero<<<blocks((long long)N * 64, TB), TB, 0, stream>>>(agg2, (long long)N * 64);
    gat_gemm_wmma<64><<<mStripes, 32 * (64 / 16), 0, stream>>>(agg1, W2, hlin2, N);
    gat_node_prep<<<blocks((long long)N * 2, TB), TB, 0, stream>>>(
        hlin2, as2, ad2, asb, adb, mbuf, sbuf, N, 32);
    gat_edge_logits<<<edgeBlk, TB, 0, stream>>>(ei, asb, adb, ebuf, mbuf, E, N);
    gat_edge_exp<<<edgeBlk, TB, 0, stream>>>(ei, ebuf, mbuf, sbuf, E, N);
    gat_edge_agg<<<blocks((long long)Etot * 64, TB), TB, 0, stream>>>(
        ei, ebuf, sbuf, hlin2, agg2, E, N, 6 /*log2(64)*/, 5 /*log2(32)*/);

    // ---------------- pooling + log_softmax --------------------------------
    gat_zero<<<blocks((long long)G * 33, TB), TB, 0, stream>>>(psum, (long long)G * 33);
    gat_finalize_pool<<<blocks((long long)N * 32, TB), TB, 0, stream>>>(
        agg2, b2, g2, be2, mu2, var2, batch, psum, pcnt, N);
    gat_pool_logsoftmax<<<G, 32, 0, stream>>>(psum, pcnt, (float*)d_out, G);
}